// GAT_24919400251446
// MI455X (gfx1250) — compile-verified
//
#include <hip/hip_runtime.h>

typedef __attribute__((ext_vector_type(2))) float v2f;
typedef __attribute__((ext_vector_type(8))) float v8f;

#define NEG_SLOPE 0.2f
#define FIN 23
#define FOUT 512   // H1*D1 = 8*64

// ---------------- helpers ----------------

// Monotonic float atomic-max via signed/unsigned integer atomics.
// Works for any mix of signs when the cell is initialized to a very negative float.
__device__ __forceinline__ void atomicMaxF(float* addr, float val) {
    if (val >= 0.0f) {
        atomicMax((int*)addr, __float_as_int(val));
    } else {
        atomicMin((unsigned int*)addr, __float_as_uint(val));
    }
}

__global__ void fill_kernel(float* __restrict__ p, long long n, float v) {
    long long i = (long long)blockIdx.x * 256 + threadIdx.x;
    if (i < n) p[i] = v;
}

__global__ void fill_ptr_kernel(float* __restrict__ p, long long n, const float* __restrict__ v) {
    long long i = (long long)blockIdx.x * 256 + threadIdx.x;
    if (i < n) p[i] = v[0];
}

// out1[i, f] = bias1[f]  (accumulator base: segment_sum + bias)
__global__ void init_out1_kernel(float* __restrict__ out1, const float* __restrict__ bias, int Nn) {
    long long i = (long long)blockIdx.x * 256 + threadIdx.x;
    if (i < (long long)Nn * FOUT) out1[i] = bias[i & (FOUT - 1)];
}

// ---------------- layer-1 dense transforms via f32 WMMA ----------------
// xl = x@Wl + bl ; xr = x@Wr + br.  One wave per 16x16 output tile,
// K = 23 padded to 24 (6 steps of V_WMMA_F32_16X16X4_F32).
// A layout (16x4 f32): lane half h, VGPR v -> K = 2h+v, M = lane%16.
// B layout (4x16 f32): lane half h, VGPR v -> K = 2h+v, N = lane%16.
// C/D layout (16x16 f32): VGPR r, half h -> M = r+8h, N = lane%16.
__global__ __launch_bounds__(128) void gemm1_kernel(
    const float* __restrict__ x,                      // [Nn, 23]
    const float* __restrict__ Wl, const float* __restrict__ bl,
    const float* __restrict__ Wr, const float* __restrict__ br,
    float* __restrict__ xl, float* __restrict__ xr,   // [Nn, 512]
    int Nn)
{
    const int lane = threadIdx.x & 31;
    const int wave = threadIdx.x >> 5;
    const int tilesM = Nn >> 4;                 // Nn multiple of 16 (50000 = 3125*16)
    const int tile = blockIdx.x * 4 + wave;
    if (tile >= tilesM * 32) return;            // wave-uniform
    const int tm = tile >> 5;                   // tiles of 16 rows
    const int tn = tile & 31;                   // 32 tiles of 16 cols
    const int m0 = tm << 4;
    const int n0 = tn << 4;
    const int h  = lane >> 4;                   // lane half
    const int r  = lane & 15;

    v8f cl = {}; v8f cr = {};
    const float* xrow = x + (long long)(m0 + r) * FIN;

    #pragma unroll
    for (int kk = 0; kk < 24; kk += 4) {
        const int k0 = kk + 2 * h;
        const int k1 = k0 + 1;
        v2f a, bL, bR;
        a.x  = (k0 < FIN) ? xrow[k0] : 0.0f;
        a.y  = (k1 < FIN) ? xrow[k1] : 0.0f;
        bL.x = (k0 < FIN) ? Wl[k0 * FOUT + n0 + r] : 0.0f;
        bL.y = (k1 < FIN) ? Wl[k1 * FOUT + n0 + r] : 0.0f;
        bR.x = (k0 < FIN) ? Wr[k0 * FOUT + n0 + r] : 0.0f;
        bR.y = (k1 < FIN) ? Wr[k1 * FOUT + n0 + r] : 0.0f;
        cl = __builtin_amdgcn_wmma_f32_16x16x4_f32(false, a, false, bL, (short)0, cl, false, false);
        cr = __builtin_amdgcn_wmma_f32_16x16x4_f32(false, a, false, bR, (short)0, cr, false, false);
    }

    const float biasL = bl[n0 + r];
    const float biasR = br[n0 + r];
    #pragma unroll
    for (int i = 0; i < 8; ++i) {
        const int row = m0 + i + 8 * h;
        xl[(long long)row * FOUT + n0 + r] = cl[i] + biasL;
        xr[(long long)row * FOUT + n0 + r] = cr[i] + biasR;
    }
}

// ---------------- layer-1 edge passes ----------------
// Pass A: logit[e,h] = sum_d att[h,d]*leakyrelu(xl[src]+xr[dst]); segment max.
// One wave per edge; lane covers 16 contiguous feature dims (head = lane/4).
__global__ __launch_bounds__(256) void logits1_kernel(
    const int* __restrict__ ei, int E, int Nn,
    const float* __restrict__ xl, const float* __restrict__ xr,
    const float* __restrict__ att,        // [8*64] flat
    float* __restrict__ logit,            // [Etot, 8]
    float* __restrict__ lmax)             // [Nn, 8]
{
    const int lane = threadIdx.x & 31;
    const long long e = (long long)blockIdx.x * 8 + (threadIdx.x >> 5);
    if (e >= (long long)E + Nn) return;
    const int src = (e < E) ? ei[e] : (int)(e - E);
    const int dst = (e < E) ? ei[E + e] : src;

    const float* pl = xl + (long long)src * FOUT + lane * 16;
    const float* pr = xr + (long long)dst * FOUT + lane * 16;
    const float* pa = att + lane * 16;
    float s = 0.0f;
    #pragma unroll
    for (int t = 0; t < 16; ++t) {
        float v = pl[t] + pr[t];
        v = (v > 0.0f) ? v : NEG_SLOPE * v;
        s += pa[t] * v;
    }
    // reduce 4 lanes belonging to the same head
    s += __shfl_xor(s, 1, 32);
    s += __shfl_xor(s, 2, 32);
    if ((lane & 3) == 0) {
        const int hd = lane >> 2;
        logit[e * 8 + hd] = s;
        atomicMaxF(&lmax[(long long)dst * 8 + hd], s);
    }
}

// Pass B: ex = exp(logit - lmax[dst]); segment sum. One thread per (edge, head).
__global__ void exp1_kernel(
    const int* __restrict__ ei, int E, int Nn,
    float* __restrict__ ebuf, const float* __restrict__ lmax,
    float* __restrict__ denom)
{
    const long long idx = (long long)blockIdx.x * 256 + threadIdx.x;
    const long long tot = ((long long)E + Nn) * 8;
    if (idx >= tot) return;
    const long long e = idx >> 3;
    const int hd = (int)(idx & 7);
    const int dst = (e < E) ? ei[E + e] : (int)(e - E);
    const float ex = expf(ebuf[idx] - lmax[(long long)dst * 8 + hd]);
    ebuf[idx] = ex;
    atomicAdd(&denom[(long long)dst * 8 + hd], ex);
}

// Pass C: out1[dst] += (ex/denom[dst]) * xl[src]. One wave per edge.
__global__ __launch_bounds__(256) void agg1_kernel(
    const int* __restrict__ ei, int E, int Nn,
    const float* __restrict__ xl, const float* __restrict__ ebuf,
    const float* __restrict__ denom, float* __restrict__ out1)
{
    const int lane = threadIdx.x & 31;
    const long long e = (long long)blockIdx.x * 8 + (threadIdx.x >> 5);
    if (e >= (long long)E + Nn) return;
    const int src = (e < E) ? ei[e] : (int)(e - E);
    const int dst = (e < E) ? ei[E + e] : src;
    const int hd = lane >> 2;
    const float alpha = ebuf[e * 8 + hd] / denom[(long long)dst * 8 + hd];
    const float* pl = xl + (long long)src * FOUT + lane * 16;
    float* po = out1 + (long long)dst * FOUT + lane * 16;
    #pragma unroll
    for (int t = 0; t < 16; ++t)
        atomicAdd(&po[t], alpha * pl[t]);
}

// ---------------- layer-2 ----------------
// xl2[i] = relu(h1[i,:]).Wl2 + bl2 ; xr2 analog. One wave per node.
__global__ __launch_bounds__(256) void l2x_kernel(
    const float* __restrict__ out1,
    const float* __restrict__ Wl2, const float* __restrict__ Wr2,
    const float* __restrict__ bl2, const float* __restrict__ br2,
    int Nn, float* __restrict__ xl2, float* __restrict__ xr2)
{
    const int lane = threadIdx.x & 31;
    const int i = blockIdx.x * 8 + (threadIdx.x >> 5);
    if (i >= Nn) return;
    const float* p = out1 + (long long)i * FOUT + lane * 16;
    float sl = 0.0f, sr = 0.0f;
    #pragma unroll
    for (int t = 0; t < 16; ++t) {
        float v = p[t];
        v = (v > 0.0f) ? v : 0.0f;            // relu between layers
        sl += v * Wl2[lane * 16 + t];
        sr += v * Wr2[lane * 16 + t];
    }
    #pragma unroll
    for (int off = 16; off; off >>= 1) {
        sl += __shfl_xor(sl, off, 32);
        sr += __shfl_xor(sr, off, 32);
    }
    if (lane == 0) {
        xl2[i] = sl + bl2[0];
        xr2[i] = sr + br2[0];
    }
}

__global__ void logits2_kernel(
    const int* __restrict__ ei, int E, int Nn,
    const float* __restrict__ xl2, const float* __restrict__ xr2,
    const float* __restrict__ att2,
    float* __restrict__ logit2, float* __restrict__ lmax2)
{
    const long long e = (long long)blockIdx.x * 256 + threadIdx.x;
    if (e >= (long long)E + Nn) return;
    const int src = (e < E) ? ei[e] : (int)(e - E);
    const int dst = (e < E) ? ei[E + e] : src;
    float v = xl2[src] + xr2[dst];
    v = (v > 0.0f) ? v : NEG_SLOPE * v;
    const float lg = v * att2[0];
    logit2[e] = lg;
    atomicMaxF(&lmax2[dst], lg);
}

__global__ void exp2_kernel(
    const int* __restrict__ ei, int E, int Nn,
    float* __restrict__ e2buf, const float* __restrict__ lmax2,
    float* __restrict__ denom2)
{
    const long long e = (long long)blockIdx.x * 256 + threadIdx.x;
    if (e >= (long long)E + Nn) return;
    const int dst = (e < E) ? ei[E + e] : (int)(e - E);
    const float ex = expf(e2buf[e] - lmax2[dst]);
    e2buf[e] = ex;
    atomicAdd(&denom2[dst], ex);
}

// alpha -> out[Nn + e]; h2[dst] += alpha * xl2[src] (out pre-filled with bias2)
__global__ void final2_kernel(
    const int* __restrict__ ei, int E, int Nn,
    const float* __restrict__ xl2, const float* __restrict__ e2buf,
    const float* __restrict__ denom2, float* __restrict__ out)
{
    const long long e = (long long)blockIdx.x * 256 + threadIdx.x;
    if (e >= (long long)E + Nn) return;
    const int src = (e < E) ? ei[e] : (int)(e - E);
    const int dst = (e < E) ? ei[E + e] : src;
    const float a = e2buf[e] / denom2[dst];
    out[Nn + e] = a;
    atomicAdd(&out[dst], a * xl2[src]);
}

// ---------------- launch ----------------

extern "C" void kernel_launch(void* const* d_in, const int* in_sizes, int n_in,
                              void* d_out, int out_size, void* d_ws, size_t ws_size,
                              hipStream_t stream) {
    const float* x     = (const float*)d_in[0];
    const int*   ei    = (const int*)d_in[1];
    const float* Wl1   = (const float*)d_in[2];
    const float* bl1   = (const float*)d_in[3];
    const float* Wr1   = (const float*)d_in[4];
    const float* br1   = (const float*)d_in[5];
    const float* att1  = (const float*)d_in[6];
    const float* bias1 = (const float*)d_in[7];
    const float* Wl2   = (const float*)d_in[8];
    const float* bl2   = (const float*)d_in[9];
    const float* Wr2   = (const float*)d_in[10];
    const float* br2   = (const float*)d_in[11];
    const float* att2  = (const float*)d_in[12];
    const float* bias2 = (const float*)d_in[13];

    const int Nn = in_sizes[0] / FIN;      // 50000
    const int E  = in_sizes[1] / 2;        // 400000
    const long long Etot = (long long)E + Nn;

    // workspace carve-up (floats)
    float* ws    = (float*)d_ws;
    float* xl1   = ws;                                    // Nn*512
    float* xr1   = xl1  + (size_t)Nn * FOUT;              // Nn*512
    float* out1  = xr1  + (size_t)Nn * FOUT;              // Nn*512
    float* ebuf  = out1 + (size_t)Nn * FOUT;              // Etot*8
    float* lmax1 = ebuf + (size_t)Etot * 8;               // Nn*8
    float* den1  = lmax1 + (size_t)Nn * 8;                // Nn*8
    float* xl2   = den1  + (size_t)Nn * 8;                // Nn
    float* xr2   = xl2 + Nn;                              // Nn
    float* lmax2 = xr2 + Nn;                              // Nn
    float* den2  = lmax2 + Nn;                            // Nn
    float* e2buf = den2 + Nn;                             // Etot
    float* out   = (float*)d_out;                         // [Nn] h, then [Etot] alpha

    auto blocks = [](long long n, int per) -> unsigned {
        return (unsigned)((n + per - 1) / per);
    };

    // init accumulators / stats (workspace is poisoned by harness)
    fill_kernel<<<blocks((long long)Nn * 8, 256), 256, 0, stream>>>(lmax1, (long long)Nn * 8, -3.0e38f);
    fill_kernel<<<blocks((long long)Nn * 8, 256), 256, 0, stream>>>(den1, (long long)Nn * 8, 0.0f);
    init_out1_kernel<<<blocks((long long)Nn * FOUT, 256), 256, 0, stream>>>(out1, bias1, Nn);
    fill_kernel<<<blocks(Nn, 256), 256, 0, stream>>>(lmax2, Nn, -3.0e38f);
    fill_kernel<<<blocks(Nn, 256), 256, 0, stream>>>(den2, Nn, 0.0f);
    fill_ptr_kernel<<<blocks(Nn, 256), 256, 0, stream>>>(out, Nn, bias2);

    // layer-1 dense transforms (WMMA f32 16x16x4)
    const int tiles = (Nn / 16) * 32;
    gemm1_kernel<<<blocks(tiles, 4), 128, 0, stream>>>(x, Wl1, bl1, Wr1, br1, xl1, xr1, Nn);

    // layer-1 attention
    logits1_kernel<<<blocks(Etot, 8), 256, 0, stream>>>(ei, E, Nn, xl1, xr1, att1, ebuf, lmax1);
    exp1_kernel<<<blocks(Etot * 8, 256), 256, 0, stream>>>(ei, E, Nn, ebuf, lmax1, den1);
    agg1_kernel<<<blocks(Etot, 8), 256, 0, stream>>>(ei, E, Nn, xl1, ebuf, den1, out1);

    // layer-2
    l2x_kernel<<<blocks(Nn, 8), 256, 0, stream>>>(out1, Wl2, Wr2, bl2, br2, Nn, xl2, xr2);
    logits2_kernel<<<blocks(Etot, 256), 256, 0, stream>>>(ei, E, Nn, xl2, xr2, att2, e2buf, lmax2);
    exp2_kernel<<<blocks(Etot, 256), 256, 0, stream>>>(ei, E, Nn, e2buf, lmax2, den2);
    final2_kernel<<<blocks(Etot, 256), 256, 0, stream>>>(ei, E, Nn, xl2, e2buf, den2, out);
}